// QuantumLayer_52587579572943
// MI455X (gfx1250) — compile-verified
//
#include <hip/hip_runtime.h>
#include <math.h>

// ---------------------------------------------------------------------------
// 16-qubit, 32-batch quantum circuit simulator for MI455X (gfx1250, wave32).
//
// Structure exploited:
//  * state before first CNOT ring is a product state -> two 256-entry complex
//    tables per batch element (no gate passes at all for embedding+H+layer0).
//  * CNOT rings are GF(2)-linear index permutations -> fused into the gather
//    of pass A (closed form) and into the measurement of pass B (XOR LUTs).
//  * layer-1 Rot gates grouped 4 qubits at a time -> 16x16 complex unitaries
//    applied as complex GEMMs built from V_WMMA_F32_16X16X4_F32.
//  * pass B's 32KB state staging tile is loaded by the Tensor Data Mover
//    (tensor_load_to_lds + s_wait_tensorcnt) instead of per-lane VMEM.
// Global traffic: one 16MB state write + one 16MB (TDM) read.
// ---------------------------------------------------------------------------

typedef __attribute__((ext_vector_type(2))) float v2f;   // A/B operand: 16x4 / 4x16 f32 = 2 VGPRs
typedef __attribute__((ext_vector_type(8))) float v8f;   // C/D operand: 16x16 f32 = 8 VGPRs
typedef __attribute__((ext_vector_type(4))) unsigned int v4u;
typedef __attribute__((ext_vector_type(8))) int          v8i;
typedef __attribute__((ext_vector_type(4))) int          v4i;

#define WMMA4(A, B, C) \
    __builtin_amdgcn_wmma_f32_16x16x4_f32(false, (A), false, (B), (short)0, (C), false, false)

struct cplx { float re, im; };

__device__ __forceinline__ cplx cmul(cplx a, cplx b) {
    return cplx{ a.re * b.re - a.im * b.im, a.re * b.im + a.im * b.re };
}

// PennyLane Rot(phi, theta, omega) = RZ(omega) RY(theta) RZ(phi)
// Args are bounded (|x| <= 2*pi): hardware V_SIN/V_COS via __sincosf is plenty.
__device__ __forceinline__ void rot_mat(float phi, float theta, float omega, cplx R[2][2]) {
    float st, ct; __sincosf(0.5f * theta, &st, &ct);
    float sp, cp; __sincosf(0.5f * (phi + omega), &sp, &cp);
    float sm, cm; __sincosf(0.5f * (phi - omega), &sm, &cm);
    R[0][0] = cplx{  cp * ct, -sp * ct };   //  e^{-i(phi+omega)/2} cos
    R[0][1] = cplx{ -cm * st, -sm * st };   // -e^{+i(phi-omega)/2} sin
    R[1][0] = cplx{  cm * st, -sm * st };   //  e^{-i(phi-omega)/2} sin
    R[1][1] = cplx{  cp * ct,  sp * ct };   //  e^{+i(phi+omega)/2} cos
}

// Element [row][col] of the 16x16 group unitary = tensor product of 4 Rot's
__device__ __forceinline__ cplx group_elem(const cplx R[4][2][2], int row, int col) {
    cplx v = R[0][row & 1][col & 1];
    v = cmul(v, R[1][(row >> 1) & 1][(col >> 1) & 1]);
    v = cmul(v, R[2][(row >> 2) & 1][(col >> 2) & 1]);
    v = cmul(v, R[3][(row >> 3) & 1][(col >> 3) & 1]);
    return v;
}

// Gather map of the r=1 CNOT ring: new[j] = old[perm1_src(j)]
__device__ __forceinline__ unsigned perm1_src(unsigned j) {
    return (j ^ (j << 1) ^ (((j >> 15) & 1u) * 3u)) & 0xFFFFu;
}

// Forward label map of the r=2 CNOT ring: |b> -> |f2(b)>
__device__ __forceinline__ unsigned f2_fwd(unsigned b) {
#pragma unroll
    for (int c = 0; c < 16; ++c) {
        unsigned t = (unsigned)((c + 2) & 15);
        b ^= ((b >> c) & 1u) << t;
    }
    return b & 0xFFFFu;
}

// Per-lane WMMA operand constants for one complex GEMM pair:
//   A-layout elements of group(RA)          : A[M=ln][K]
//   B-layout elements of group(RB)-transpose: B[K][N=ln] = RB[ln][K]
__device__ __forceinline__ void make_ab_consts(const cplx RA[4][2][2], const cplx RB[4][2][2],
                                               int ln, int hi,
                                               v2f ar[4], v2f ai[4], v2f ani[4],
                                               v2f br[4], v2f bi[4], v2f bni[4]) {
#pragma unroll
    for (int kc = 0; kc < 4; ++kc) {
#pragma unroll
        for (int v = 0; v < 2; ++v) {
            int K = 4 * kc + v + 2 * hi;
            cplx ua = group_elem(RA, ln, K);
            ar[kc][v] = ua.re; ai[kc][v] = ua.im; ani[kc][v] = -ua.im;
            cplx ub = group_elem(RB, ln, K);
            br[kc][v] = ub.re; bi[kc][v] = ub.im; bni[kc][v] = -ub.im;
        }
    }
}

// ---------------------------------------------------------------------------
// K1: per-batch product-state tables.
//   u_w = Rot_l0(w) * H * RY(x_w) |0>;  Alow[t] = prod_{w<8} u_w[bit],  Ahigh likewise.
// ---------------------------------------------------------------------------
__global__ __launch_bounds__(256) void qsim_tables(const float* __restrict__ inputs,
                                                   const float* __restrict__ weights,
                                                   float2* __restrict__ Alow,
                                                   float2* __restrict__ Ahigh) {
    __shared__ float2 u[16][2];
    const int k = blockIdx.x;
    const int t = threadIdx.x;

    if (t < 16) {
        const int w = t;
        float x = inputs[k * 16 + w];
        float sh, ch; __sincosf(0.5f * x, &sh, &ch);
        const float inv_sqrt2 = 0.70710678118654752f;
        float p0 = (ch + sh) * inv_sqrt2;   // H * RY(x)|0>, real
        float p1 = (ch - sh) * inv_sqrt2;
        const float* wp = weights + w * 3;  // layer 0
        cplx R[2][2]; rot_mat(wp[0], wp[1], wp[2], R);
        cplx u0 = cplx{ R[0][0].re * p0 + R[0][1].re * p1, R[0][0].im * p0 + R[0][1].im * p1 };
        cplx u1 = cplx{ R[1][0].re * p0 + R[1][1].re * p1, R[1][0].im * p0 + R[1][1].im * p1 };
        u[w][0] = make_float2(u0.re, u0.im);
        u[w][1] = make_float2(u1.re, u1.im);
    }
    __syncthreads();

    cplx lo = cplx{1.0f, 0.0f}, hi = cplx{1.0f, 0.0f};
#pragma unroll
    for (int i = 0; i < 8; ++i) {
        int bit = (t >> i) & 1;
        float2 a = u[i][bit];     lo = cmul(lo, cplx{a.x, a.y});
        float2 b = u[8 + i][bit]; hi = cmul(hi, cplx{b.x, b.y});
    }
    Alow[k * 256 + t]  = make_float2(lo.re, lo.im);
    Ahigh[k * 256 + t] = make_float2(hi.re, hi.im);
}

// ---------------------------------------------------------------------------
// K2 (pass A): fuse {product-state eval + r=1 CNOT permutation} into the tile
// gather, then apply layer-1 Rot groups 0 (bits 0-3) and 1 (bits 4-7):
//   Out = U1 x T x U0^T   per 256-amplitude tile (h = bits 8-15 fixed).
// One wave per tile, 8 tiles per 256-thread block.
// ---------------------------------------------------------------------------
__global__ __launch_bounds__(256) void qsim_pass_a(const float* __restrict__ weights,
                                                   const float2* __restrict__ Alow,
                                                   const float2* __restrict__ Ahigh,
                                                   float2* __restrict__ S1) {
    __shared__ float2 lds_lo[256];
    __shared__ float2 lds_hi[256];
    __shared__ float  psc[8][2][16][16];   // per-wave D->A relayout scratch

    const int k    = blockIdx.y;
    const int t    = threadIdx.x;
    const int lane = t & 31;
    const int wave = t >> 5;
    const int hi   = (lane >> 4) & 1;
    const int ln   = lane & 15;

    lds_lo[t] = Alow[k * 256 + t];
    lds_hi[t] = Ahigh[k * 256 + t];

    // layer-1 Rot gates, wires 0..3 (U0) and 4..7 (U1)
    cplx R0[4][2][2], R1[4][2][2];
#pragma unroll
    for (int i = 0; i < 4; ++i) {
        const float* w0 = weights + (16 + i) * 3;     rot_mat(w0[0], w0[1], w0[2], R0[i]);
        const float* w1 = weights + (16 + 4 + i) * 3; rot_mat(w1[0], w1[1], w1[2], R1[i]);
    }
    v2f a1r[4], a1i[4], a1ni[4], b0r[4], b0i[4], b0ni[4];
    make_ab_consts(R1, R0, ln, hi, a1r, a1i, a1ni, b0r, b0i, b0ni);
    __syncthreads();

    const int h = blockIdx.x * 8 + wave;

    // Build T in B layout: T[K=bits4-7][N=bits0-3], value via perm1 + tables
    v2f tr[4], ti[4];
#pragma unroll
    for (int kc = 0; kc < 4; ++kc) {
#pragma unroll
        for (int v = 0; v < 2; ++v) {
            int K = 4 * kc + v + 2 * hi;
            unsigned j = (unsigned)(h * 256 + K * 16 + ln);
            unsigned src = perm1_src(j);
            float2 lo = lds_lo[src & 255u];
            float2 hg = lds_hi[src >> 8];
            cplx val = cmul(cplx{lo.x, lo.y}, cplx{hg.x, hg.y});
            tr[kc][v] = val.re; ti[kc][v] = val.im;
        }
    }

    // step 1: P = U1 x T   (complex; minus folded into pre-negated a1ni)
    v8f pr = {0.f,0.f,0.f,0.f,0.f,0.f,0.f,0.f};
    v8f pi = {0.f,0.f,0.f,0.f,0.f,0.f,0.f,0.f};
#pragma unroll
    for (int kc = 0; kc < 4; ++kc) {
        pr = WMMA4(a1r[kc],  tr[kc], pr);
        pr = WMMA4(a1ni[kc], ti[kc], pr);
        pi = WMMA4(a1r[kc],  ti[kc], pi);
        pi = WMMA4(a1i[kc],  tr[kc], pi);
    }

    // D layout -> LDS -> A layout
#pragma unroll
    for (int jj = 0; jj < 8; ++jj) {
        int M = jj + 8 * hi;
        psc[wave][0][M][ln] = pr[jj];
        psc[wave][1][M][ln] = pi[jj];
    }
    __syncthreads();
    v2f par[4], pai[4];
#pragma unroll
    for (int kc = 0; kc < 4; ++kc) {
#pragma unroll
        for (int v = 0; v < 2; ++v) {
            int K = 4 * kc + v + 2 * hi;
            par[kc][v] = psc[wave][0][ln][K];
            pai[kc][v] = psc[wave][1][ln][K];
        }
    }
    __syncthreads();

    // step 2: Out = P x U0^T
    v8f dr = {0.f,0.f,0.f,0.f,0.f,0.f,0.f,0.f};
    v8f di = {0.f,0.f,0.f,0.f,0.f,0.f,0.f,0.f};
#pragma unroll
    for (int kc = 0; kc < 4; ++kc) {
        dr = WMMA4(par[kc], b0r[kc],  dr);
        dr = WMMA4(pai[kc], b0ni[kc], dr);
        di = WMMA4(par[kc], b0i[kc],  di);
        di = WMMA4(pai[kc], b0r[kc],  di);
    }

    // store (lanes 0-15 write one 128B row, lanes 16-31 the row+8: coalesced)
    float2* out = S1 + (size_t)k * 65536 + (size_t)h * 256;
#pragma unroll
    for (int jj = 0; jj < 8; ++jj) {
        int M = jj + 8 * hi;
        out[M * 16 + ln] = make_float2(dr[jj], di[jj]);
    }
}

// ---------------------------------------------------------------------------
// K3 (pass B): apply layer-1 Rot groups 2 (bits 8-11) and 3 (bits 12-15):
//   Out = U3 x T x U2^T  per tile (l = bits 0-7 fixed), then fuse the r=2
// CNOT permutation + <Z_w> accumulation. The 256(H) x 16(l) staging tile is
// pulled into LDS by the Tensor Data Mover (one descriptor per workgroup).
// Deterministic tree reduction -> per-block partials.
// ---------------------------------------------------------------------------
__global__ __launch_bounds__(256) void qsim_pass_b(const float* __restrict__ weights,
                                                   const float2* __restrict__ S1,
                                                   float* __restrict__ partial) {
    __shared__ float2 stage[4096];                  // 32 KB: 256 H x 16 l, row-major [H][li]
    __shared__ float  psc[8][2][16][16];            // 16 KB
    __shared__ unsigned short F2l[256], F2h[256];   // r=2 ring forward map LUTs

    const int k     = blockIdx.y;
    const int lbase = blockIdx.x * 16;
    const int t     = threadIdx.x;
    const int lane  = t & 31;
    const int wave  = t >> 5;
    const int hi    = (lane >> 4) & 1;
    const int ln    = lane & 15;

    F2l[t] = (unsigned short)f2_fwd((unsigned)t);
    F2h[t] = (unsigned short)f2_fwd((unsigned)t << 8);

    // layer-1 Rot gates, wires 8..11 (U2) and 12..15 (U3)
    cplx R2[4][2][2], R3[4][2][2];
#pragma unroll
    for (int i = 0; i < 4; ++i) {
        const float* w2 = weights + (16 + 8 + i) * 3;  rot_mat(w2[0], w2[1], w2[2], R2[i]);
        const float* w3 = weights + (16 + 12 + i) * 3; rot_mat(w3[0], w3[1], w3[2], R3[i]);
    }
    v2f a3r[4], a3i[4], a3ni[4], b2r[4], b2i[4], b2ni[4];
    make_ab_consts(R3, R2, ln, hi, a3r, a3i, a3ni, b2r, b2i, b2ni);

    // --- TDM: stage[H*16+li] = S1[k*65536 + H*256 + lbase + li] ----------------
    // 2D descriptor: data_size=8B (code 3), tile_dim0=16 (contiguous l),
    // tile_dim1=256 (H rows), tensor_dim0_stride=256 elements. LDS dest is the
    // contiguous row-major tile order, exactly matching stage[][].
    if (wave == 0) {
        const float2* gsrc = S1 + (size_t)k * 65536 + lbase;
        unsigned long long ga = (unsigned long long)(size_t)gsrc;
        unsigned lds_off = (unsigned)(size_t)(void*)&stage[0];

        v4u g0 = { 0x1u,                                   // count=1, is_restore=0, gather=0
                   lds_off,                                // lds_addr  (bits 63:32)
                   (unsigned)(ga & 0xFFFFFFFFu),           // global_addr[31:0]
                   (unsigned)(((ga >> 32) & 0x01FFFFFFu)   // global_addr[56:32]
                              | 0x80000000u) };            // type=2 (bits 127:126)
        v8i g1 = { (int)(3u << 16),        // data_size=3 (8B); wg_mask=0; no flags
                   (int)(16u << 16),       // tensor_dim0[15:0]=16  (bits 63:48)
                   (int)(256u << 16),      // tensor_dim0[31:16]=0, tensor_dim1[15:0]=256
                   (int)(16u << 16),       // tensor_dim1[31:16]=0, tile_dim0=16
                   256,                    // tile_dim1=256, tile_dim2=0
                   256,                    // tensor_dim0_stride[31:0]=256
                   0,                      // stride[47:32]=0, tensor_dim1_stride lo=0
                   0 };
        v4i gz = { 0, 0, 0, 0 };
#if __clang_major__ >= 23
        v8i gz8 = { 0, 0, 0, 0, 0, 0, 0, 0 };
        __builtin_amdgcn_tensor_load_to_lds(g0, g1, gz, gz, gz8, 0);
#else
        __builtin_amdgcn_tensor_load_to_lds(g0, g1, gz, gz, 0);
#endif
        __builtin_amdgcn_s_wait_tensorcnt(0);
    }
    __syncthreads();   // publish TDM-written LDS (and F2 LUTs) to all waves

    float zacc[16];
#pragma unroll
    for (int w = 0; w < 16; ++w) zacc[w] = 0.0f;

    for (int tt = 0; tt < 2; ++tt) {        // each wave does 2 of the 16 l's
        const int li = wave * 2 + tt;

        // T in B layout: B[K=bits12-15][N=bits8-11] = stage[(K*16+N)*16 + li]
        v2f tr[4], ti[4];
#pragma unroll
        for (int kc = 0; kc < 4; ++kc) {
#pragma unroll
            for (int v = 0; v < 2; ++v) {
                int K = 4 * kc + v + 2 * hi;
                float2 e = stage[(K * 16 + ln) * 16 + li];
                tr[kc][v] = e.x; ti[kc][v] = e.y;
            }
        }

        // step 1: P = U3 x T
        v8f pr = {0.f,0.f,0.f,0.f,0.f,0.f,0.f,0.f};
        v8f pi = {0.f,0.f,0.f,0.f,0.f,0.f,0.f,0.f};
#pragma unroll
        for (int kc = 0; kc < 4; ++kc) {
            pr = WMMA4(a3r[kc],  tr[kc], pr);
            pr = WMMA4(a3ni[kc], ti[kc], pr);
            pi = WMMA4(a3r[kc],  ti[kc], pi);
            pi = WMMA4(a3i[kc],  tr[kc], pi);
        }

        // D -> A relayout
#pragma unroll
        for (int jj = 0; jj < 8; ++jj) {
            int M = jj + 8 * hi;
            psc[wave][0][M][ln] = pr[jj];
            psc[wave][1][M][ln] = pi[jj];
        }
        __syncthreads();
        v2f par[4], pai[4];
#pragma unroll
        for (int kc = 0; kc < 4; ++kc) {
#pragma unroll
            for (int v = 0; v < 2; ++v) {
                int K = 4 * kc + v + 2 * hi;
                par[kc][v] = psc[wave][0][ln][K];
                pai[kc][v] = psc[wave][1][ln][K];
            }
        }
        __syncthreads();

        // step 2: Out = P x U2^T
        v8f dr = {0.f,0.f,0.f,0.f,0.f,0.f,0.f,0.f};
        v8f di = {0.f,0.f,0.f,0.f,0.f,0.f,0.f,0.f};
#pragma unroll
        for (int kc = 0; kc < 4; ++kc) {
            dr = WMMA4(par[kc], b2r[kc],  dr);
            dr = WMMA4(pai[kc], b2ni[kc], dr);
            di = WMMA4(par[kc], b2i[kc],  di);
            di = WMMA4(pai[kc], b2r[kc],  di);
        }

        // fused measurement through the r=2 permutation:
        // <Z_w> += |s2[b]|^2 * sign(bit_w(f2(b)))
        unsigned f2lo = (unsigned)F2l[lbase + li];
#pragma unroll
        for (int jj = 0; jj < 8; ++jj) {
            int M = jj + 8 * hi;
            int H = M * 16 + ln;
            float p = dr[jj] * dr[jj] + di[jj] * di[jj];
            unsigned f2b = f2lo ^ (unsigned)F2h[H];
#pragma unroll
            for (int w = 0; w < 16; ++w)
                zacc[w] += ((f2b >> w) & 1u) ? -p : p;
        }
    }

    // deterministic workgroup tree reduction (reuse stage as float scratch)
    __syncthreads();
    float* red = (float*)stage;             // 256*16 floats = 16 KB
#pragma unroll
    for (int w = 0; w < 16; ++w) red[t * 16 + w] = zacc[w];
    __syncthreads();
    for (int off = 128; off > 0; off >>= 1) {
        if (t < off) {
#pragma unroll
            for (int w = 0; w < 16; ++w)
                red[t * 16 + w] += red[(t + off) * 16 + w];
        }
        __syncthreads();
    }
    if (t < 16)
        partial[(size_t)(k * 16 + blockIdx.x) * 16 + t] = red[t];
}

// ---------------------------------------------------------------------------
// K4: deterministic final reduction over the 16 l-blocks per batch element.
// ---------------------------------------------------------------------------
__global__ __launch_bounds__(512) void qsim_reduce(const float* __restrict__ partial,
                                                   float* __restrict__ out) {
    const int i = threadIdx.x;              // 512 = 32 batch * 16 wires
    const int k = i >> 4;
    const int w = i & 15;
    float s = 0.0f;
#pragma unroll
    for (int lb = 0; lb < 16; ++lb)
        s += partial[(size_t)(k * 16 + lb) * 16 + w];
    out[k * 16 + w] = s;
}

// ---------------------------------------------------------------------------
// launch
// ---------------------------------------------------------------------------
extern "C" void kernel_launch(void* const* d_in, const int* in_sizes, int n_in,
                              void* d_out, int out_size, void* d_ws, size_t ws_size,
                              hipStream_t stream) {
    const float* inputs  = (const float*)d_in[0];   // (32, 16) f32
    const float* weights = (const float*)d_in[1];   // (2, 16, 3) f32

    char* ws = (char*)d_ws;
    float2* Alow    = (float2*)(ws);                                  // 64 KB
    float2* Ahigh   = (float2*)(ws + 65536);                          // 64 KB
    float2* S1      = (float2*)(ws + 131072);                         // 16 MB state
    float*  partial = (float*)(ws + 131072 + (size_t)32 * 65536 * 8); // 32 KB

    qsim_tables<<<32, 256, 0, stream>>>(inputs, weights, Alow, Ahigh);
    qsim_pass_a<<<dim3(32, 32), 256, 0, stream>>>(weights, Alow, Ahigh, S1);
    qsim_pass_b<<<dim3(16, 32), 256, 0, stream>>>(weights, S1, partial);
    qsim_reduce<<<1, 512, 0, stream>>>(partial, (float*)d_out);
}